// IrrepMultiHeadAttention_64673617543729
// MI455X (gfx1250) — compile-verified
//
#include <hip/hip_runtime.h>
#include <math.h>

// ---------------------------------------------------------------------------
// IrrepMultiHeadAttention for MI455X (gfx1250, wave32, WMMA + async LDS DMA).
//
// Math restructure (generators antisymmetric => E=expm(phi.G) orthogonal):
//   kl_ij = 0.5 * ( dot(a_i, b_j) + d_j - e_i )
//     a_i = [ vec(Q_i + u_i u_i^T) (256) , u_i (16) ]           (272 floats)
//     b_j = [ vec(R_j)             (256) , -2 h_j (16) ]        (272 floats)
//     Q_i = E_i^T (Sigma_i+eps I) E_i,  u_i = E_i^T mu_i
//     M_j = (Sigma_j+eps I)^{-1},  R_j = F_j M_j F_j^T,  h_j = F_j M_j mu_j
//     d_j = mu_j^T M_j mu_j + logdet(Sigma_j+eps I),  e_i = K + logdet(...)_i
//   beta = softmax_j(-max(kl,0)); mu_agg_i = E_i (beta @ V)_i, V_j = F_j mu_j
//
// O(N^2) work -> f32 GEMMs on V_WMMA_F32_16X16X4_F32, operand tiles staged
// into LDS with GLOBAL_LOAD_ASYNC_TO_LDS_B128 + s_wait_asynccnt.
// ---------------------------------------------------------------------------

#define NDIM 256
#define BN   512      // B*N
#define KD   16
#define FEAT 272      // 256 + 16, multiple of 4 for 16x16x4 K-steps
#define EPSR 1e-6f

typedef float v2f __attribute__((ext_vector_type(2)));
typedef float v8f __attribute__((ext_vector_type(8)));

// Truncate a flat shared-aperture pointer to its 32-bit LDS byte offset.
__device__ __forceinline__ unsigned lds_off(const void* p) {
  return (unsigned)(size_t)p;
}

// One async DMA: 32 lanes x 16B = 512B global -> LDS, tracked by ASYNCcnt.
__device__ __forceinline__ void async_copy_b128(unsigned lds_addr,
                                                unsigned long long gaddr) {
  asm volatile("global_load_async_to_lds_b128 %0, %1, off"
               :: "v"(lds_addr), "v"(gaddr) : "memory");
}

__device__ __forceinline__ void wait_async0() {
  asm volatile("s_wait_asynccnt 0" ::: "memory");
}

// ---------------------------------------------------------------------------
// expm of a 16x16 matrix via scaling-and-squaring + degree-10 Horner Taylor.
// One element per thread (tid = r*16+c), matrices in LDS.
// ---------------------------------------------------------------------------
__device__ __forceinline__ void expm16(const float* M, float* T0, float* T1,
                                       float* out, int s, int tid, int r, int c) {
  float* src = T0;
  float* dst = T1;
  src[tid] = (r == c) ? 1.0f : 0.0f;
  __syncthreads();
  for (int k = 10; k >= 1; --k) {
    float acc = 0.0f;
    #pragma unroll
    for (int kk = 0; kk < 16; ++kk) acc += M[r * 16 + kk] * src[kk * 16 + c];
    acc = acc / (float)k + ((r == c) ? 1.0f : 0.0f);
    dst[tid] = acc;
    __syncthreads();
    float* t = src; src = dst; dst = t;
  }
  for (int q = 0; q < s; ++q) {
    float acc = 0.0f;
    #pragma unroll
    for (int kk = 0; kk < 16; ++kk) acc += src[r * 16 + kk] * src[kk * 16 + c];
    dst[tid] = acc;
    __syncthreads();
    float* t = src; src = dst; dst = t;
  }
  out[tid] = src[tid];
  __syncthreads();
}

// ---------------------------------------------------------------------------
// Per-(b,n) preprocessing: one 256-thread block per Gaussian.
// ---------------------------------------------------------------------------
__global__ void __launch_bounds__(256)
prep_kernel(const float* __restrict__ mu_q, const float* __restrict__ sigma_q,
            const float* __restrict__ phi, const float* __restrict__ gens,
            float* __restrict__ Af, float* __restrict__ Bf,
            float* __restrict__ dv, float* __restrict__ ev,
            float* __restrict__ Vv, float* __restrict__ Em) {
  __shared__ float bufA[256], bufB[256], bufC[256];
  __shared__ float sE[256], sF[256], sAr[256], sCh[256], sW[256], sM[256];
  __shared__ float sRed[256];
  __shared__ float sMu[16], sU[16], sVvec[16], sH[16], sTv[16];
  __shared__ float sC0;

  const int tid = threadIdx.x;
  const int r = tid >> 4, c = tid & 15;
  const int n = blockIdx.x;              // global (b*N + i)

  // X = sum_a phi_a * G_a   (antisymmetric)
  const float p0 = phi[n * 3 + 0], p1 = phi[n * 3 + 1], p2 = phi[n * 3 + 2];
  const float x = p0 * gens[r * 16 + c] + p1 * gens[256 + r * 16 + c] +
                  p2 * gens[512 + r * 16 + c];

  // Frobenius norm -> scaling exponent s (uniform across block)
  sRed[tid] = x * x;
  __syncthreads();
  for (int off = 128; off > 0; off >>= 1) {
    if (tid < off) sRed[tid] += sRed[tid + off];
    __syncthreads();
  }
  const float nrm = sqrtf(sRed[0]);
  __syncthreads();
  int s = 0; float scl = 1.0f;
  while (nrm * scl > 0.5f && s < 12) { scl *= 0.5f; ++s; }

  // E = expm(X), F = expm(-X)
  bufA[tid] = x * scl;  __syncthreads();
  expm16(bufA, bufB, bufC, sE, s, tid, r, c);
  bufA[tid] = -x * scl; __syncthreads();
  expm16(bufA, bufB, bufC, sF, s, tid, r, c);

  // A = Sigma + eps I ; in-place Cholesky (lower)
  const float areg = sigma_q[(size_t)n * 256 + tid] + ((r == c) ? EPSR : 0.0f);
  sAr[tid] = areg;
  sCh[tid] = areg;
  __syncthreads();
  for (int k = 0; k < 16; ++k) {
    if (r == k && c == k) sCh[tid] = sqrtf(sCh[tid]);
    __syncthreads();
    if (c == k && r > k) sCh[tid] /= sCh[k * 16 + k];
    __syncthreads();
    if (r > k && c > k && c <= r) sCh[tid] -= sCh[r * 16 + k] * sCh[c * 16 + k];
    __syncthreads();
  }
  // logdet = 2 * sum log diag(L)
  sRed[tid] = (r == c) ? logf(sCh[tid]) : 0.0f;
  __syncthreads();
  for (int off = 128; off > 0; off >>= 1) {
    if (tid < off) sRed[tid] += sRed[tid + off];
    __syncthreads();
  }
  const float logdet = 2.0f * sRed[0];
  __syncthreads();

  // W = L^{-1} via forward substitution (columns in parallel)
  sW[tid] = (r == c) ? 1.0f : 0.0f;
  __syncthreads();
  for (int k = 0; k < 16; ++k) {
    if (r == k) sW[tid] /= sCh[k * 16 + k];
    __syncthreads();
    if (r > k) sW[tid] -= sCh[r * 16 + k] * sW[k * 16 + c];
    __syncthreads();
  }
  // M = A^{-1} = W^T W
  {
    float acc = 0.0f;
    #pragma unroll
    for (int k = 0; k < 16; ++k) acc += sW[k * 16 + r] * sW[k * 16 + c];
    sM[tid] = acc;
  }
  if (tid < 16) sMu[tid] = mu_q[(size_t)n * 16 + tid];
  __syncthreads();

  if (tid < 16) {
    float tv = 0.0f, u = 0.0f, v = 0.0f;
    #pragma unroll
    for (int k = 0; k < 16; ++k) {
      tv += sM[tid * 16 + k] * sMu[k];   // t = M mu
      u  += sE[k * 16 + tid] * sMu[k];   // u = E^T mu
      v  += sF[tid * 16 + k] * sMu[k];   // v = F mu
    }
    sTv[tid] = tv; sU[tid] = u; sVvec[tid] = v;
  }
  __syncthreads();
  if (tid < 16) {
    float h = 0.0f;
    #pragma unroll
    for (int k = 0; k < 16; ++k) h += sF[tid * 16 + k] * sTv[k];  // h = F M mu
    sH[tid] = h;
  }
  if (tid == 0) {
    float cc = 0.0f;
    #pragma unroll
    for (int k = 0; k < 16; ++k) cc += sMu[k] * sTv[k];           // c = mu^T M mu
    sC0 = cc;
  }
  __syncthreads();

  // Q = E^T A E  (tmp = A E, Q = E^T tmp), then a-feature = Q + u u^T
  {
    float acc = 0.0f;
    #pragma unroll
    for (int k = 0; k < 16; ++k) acc += sAr[r * 16 + k] * sE[k * 16 + c];
    bufB[tid] = acc;
  }
  __syncthreads();
  {
    float q = 0.0f;
    #pragma unroll
    for (int k = 0; k < 16; ++k) q += sE[k * 16 + r] * bufB[k * 16 + c];
    Af[(size_t)n * FEAT + tid] = q + sU[r] * sU[c];
  }
  // R = F M F^T  (tmp = F M, R = tmp F^T)
  {
    float acc = 0.0f;
    #pragma unroll
    for (int k = 0; k < 16; ++k) acc += sF[r * 16 + k] * sM[k * 16 + c];
    bufC[tid] = acc;
  }
  __syncthreads();
  {
    float rv = 0.0f;
    #pragma unroll
    for (int k = 0; k < 16; ++k) rv += bufC[r * 16 + k] * sF[c * 16 + k];
    Bf[(size_t)n * FEAT + tid] = rv;
  }
  Em[(size_t)n * 256 + tid] = sE[tid];
  if (tid < 16) {
    Af[(size_t)n * FEAT + 256 + tid] = sU[tid];
    Bf[(size_t)n * FEAT + 256 + tid] = -2.0f * sH[tid];
    Vv[(size_t)n * 16 + tid] = sVvec[tid];
  }
  if (tid == 0) {
    dv[n] = sC0 + logdet;
    ev[n] = 16.0f + logdet;
  }
}

// ---------------------------------------------------------------------------
// Score GEMM: S[b] = A(256x272) @ B(256x272)^T via v_wmma_f32_16x16x4_f32.
// One wave per 16x16 output tile. Both 16x272 operand tiles are staged into
// LDS with async b128 DMA (ASYNCcnt), then the fully-unrolled K loop feeds
// WMMA from ds_load. Fused kl epilogue.
// ---------------------------------------------------------------------------
__global__ void __launch_bounds__(32)
score_gemm(const float* __restrict__ Af, const float* __restrict__ Bf,
           const float* __restrict__ dv, const float* __restrict__ ev,
           float* __restrict__ S) {
  __shared__ __align__(16) float sA[16 * FEAT];   // 17408 B
  __shared__ __align__(16) float sB[16 * FEAT];   // 17408 B
  const int blk = blockIdx.x;
  const int b  = blk >> 8;
  const int it = (blk >> 4) & 15;
  const int jt = blk & 15;
  const int lane = threadIdx.x;

  // Stage tiles: 16*272*4 = 17408 B each = 34 waves of 512 B async DMA.
  {
    const unsigned long long gA =
        (unsigned long long)(Af + (size_t)(b * NDIM + it * 16) * FEAT) + lane * 16;
    const unsigned long long gB =
        (unsigned long long)(Bf + (size_t)(b * NDIM + jt * 16) * FEAT) + lane * 16;
    const unsigned lA = lds_off(&sA[0]) + lane * 16;
    const unsigned lB = lds_off(&sB[0]) + lane * 16;
    #pragma unroll
    for (int q = 0; q < 34; ++q) {
      async_copy_b128(lA + q * 512u, gA + q * 512ull);
      async_copy_b128(lB + q * 512u, gB + q * 512ull);
    }
    wait_async0();   // single wave per block: no barrier needed
  }

  const int h = lane >> 4, m = lane & 15;
  v8f acc = {};
  #pragma unroll
  for (int k0 = 0; k0 < FEAT; k0 += 4) {
    v2f a, bb;
    a.x  = sA[m * FEAT + k0 + 2 * h];
    a.y  = sA[m * FEAT + k0 + 2 * h + 1];
    bb.x = sB[m * FEAT + k0 + 2 * h];
    bb.y = sB[m * FEAT + k0 + 2 * h + 1];
    acc = __builtin_amdgcn_wmma_f32_16x16x4_f32(false, a, false, bb,
                                                (short)0, acc, false, false);
  }

  const int j  = jt * 16 + m;
  const float dj = dv[b * NDIM + j];
  #pragma unroll
  for (int rr = 0; rr < 8; ++rr) {
    const int gi = b * NDIM + it * 16 + rr + 8 * h;   // global row
    float kl = 0.5f * (acc[rr] + dj - ev[gi]);
    kl = fmaxf(kl, 0.0f);
    S[(size_t)gi * NDIM + j] = kl;
  }
}

// ---------------------------------------------------------------------------
// Row softmax of -kl (temperature 1), in place.
// ---------------------------------------------------------------------------
__global__ void __launch_bounds__(256)
softmax_rows(float* __restrict__ S) {
  __shared__ float red[256];
  const int row = blockIdx.x;
  const int tid = threadIdx.x;
  const float x = -S[(size_t)row * NDIM + tid];
  red[tid] = x; __syncthreads();
  for (int off = 128; off > 0; off >>= 1) {
    if (tid < off) red[tid] = fmaxf(red[tid], red[tid + off]);
    __syncthreads();
  }
  const float mx = red[0];
  __syncthreads();
  const float p = __expf(x - mx);
  red[tid] = p; __syncthreads();
  for (int off = 128; off > 0; off >>= 1) {
    if (tid < off) red[tid] += red[tid + off];
    __syncthreads();
  }
  S[(size_t)row * NDIM + tid] = p * (1.0f / red[0]);
}

// ---------------------------------------------------------------------------
// Aggregation GEMM: mu_tmp[b] = beta(256x256) @ V(256x16) via WMMA.
// V (16 KB) staged into LDS via async DMA; beta rows streamed from global.
// ---------------------------------------------------------------------------
__global__ void __launch_bounds__(32)
agg_gemm(const float* __restrict__ S, const float* __restrict__ Vv,
         float* __restrict__ mt) {
  __shared__ __align__(16) float sV[NDIM * 16];    // 16384 B
  const int blk = blockIdx.x;
  const int b = blk >> 4, it = blk & 15;
  const int lane = threadIdx.x;

  {
    const unsigned long long gV =
        (unsigned long long)(Vv + (size_t)b * NDIM * 16) + lane * 16;
    const unsigned lV = lds_off(&sV[0]) + lane * 16;
    #pragma unroll
    for (int q = 0; q < 32; ++q) {     // 16384 / 512
      async_copy_b128(lV + q * 512u, gV + q * 512ull);
    }
    wait_async0();
  }

  const int h = lane >> 4, m = lane & 15;
  const float* arow = S + (size_t)(b * NDIM + it * 16 + m) * NDIM + 2 * h;

  v8f acc = {};
  #pragma unroll
  for (int k0 = 0; k0 < NDIM; k0 += 4) {
    v2f a = *(const v2f*)(arow + k0);
    v2f bb;
    bb.x = sV[(k0 + 2 * h) * 16 + m];
    bb.y = sV[(k0 + 2 * h + 1) * 16 + m];
    acc = __builtin_amdgcn_wmma_f32_16x16x4_f32(false, a, false, bb,
                                                (short)0, acc, false, false);
  }
  #pragma unroll
  for (int rr = 0; rr < 8; ++rr) {
    const int gi = b * NDIM + it * 16 + rr + 8 * h;
    mt[(size_t)gi * 16 + m] = acc[rr];
  }
}

// ---------------------------------------------------------------------------
// mu_agg_i = E_i @ mu_tmp_i (16x16 matvec per row).
// ---------------------------------------------------------------------------
__global__ void __launch_bounds__(256)
final_transform(const float* __restrict__ Em, const float* __restrict__ mt,
                float* __restrict__ out) {
  const int idx = blockIdx.x * 256 + threadIdx.x;   // < B*N*16
  const int ni = idx >> 4, kp = idx & 15;
  const float* er = Em + (size_t)ni * 256 + kp * 16;
  const float* t  = mt + (size_t)ni * 16;
  float acc = 0.0f;
  #pragma unroll
  for (int k = 0; k < 16; ++k) acc += er[k] * t[k];
  out[idx] = acc;
}

// ---------------------------------------------------------------------------
extern "C" void kernel_launch(void* const* d_in, const int* in_sizes, int n_in,
                              void* d_out, int out_size, void* d_ws, size_t ws_size,
                              hipStream_t stream) {
  (void)in_sizes; (void)n_in; (void)out_size; (void)ws_size;
  const float* mu_q    = (const float*)d_in[0];   // (B,N,K)
  const float* sigma_q = (const float*)d_in[1];   // (B,N,K,K)
  const float* phi     = (const float*)d_in[2];   // (B,N,3)
  const float* gens    = (const float*)d_in[3];   // (3,K,K)

  float* ws = (float*)d_ws;
  float* Af = ws;                       // BN*FEAT = 139264
  float* Bf = Af + (size_t)BN * FEAT;   // 139264
  float* dv = Bf + (size_t)BN * FEAT;   // 512
  float* ev = dv + BN;                  // 512
  float* Vv = ev + BN;                  // BN*16 = 8192
  float* Em = Vv + (size_t)BN * 16;     // BN*256 = 131072
  float* S  = Em + (size_t)BN * 256;    // BN*N   = 131072
  float* mt = S  + (size_t)BN * NDIM;   // BN*16  = 8192
  // total ~2.13 MB of f32 workspace

  prep_kernel<<<BN, 256, 0, stream>>>(mu_q, sigma_q, phi, gens,
                                      Af, Bf, dv, ev, Vv, Em);
  score_gemm<<<BN, 32, 0, stream>>>(Af, Bf, dv, ev, S);
  softmax_rows<<<BN, 256, 0, stream>>>(S);
  agg_gemm<<<32, 32, 0, stream>>>(S, Vv, mt);
  final_transform<<<32, 256, 0, stream>>>(Em, mt, (float*)d_out);
}